// SparseConvolutionDownsample_72043781423732
// MI455X (gfx1250) — compile-verified
//
#include <hip/hip_runtime.h>

typedef __attribute__((ext_vector_type(2))) float f32x2;
typedef __attribute__((ext_vector_type(4))) float f32x4;
typedef __attribute__((ext_vector_type(8))) float f32x8;

#define CIN  32
#define COUT 64
#define KVOL 4
#define BN_EPS 1e-4f

// ---------------------------------------------------------------------------
// Kernel 1: zero the output accumulator and the stats scratch.
// ---------------------------------------------------------------------------
__global__ void zero_kernel(float* __restrict__ out, int out_elems,
                            float* __restrict__ stats) {
  int i = blockIdx.x * blockDim.x + threadIdx.x;
  int stride = gridDim.x * blockDim.x;
  int n4 = out_elems >> 2;
  f32x4 z = {0.f, 0.f, 0.f, 0.f};
  for (int j = i; j < n4; j += stride) ((f32x4*)out)[j] = z;
  if (blockIdx.x == 0 && threadIdx.x < 2 * COUT) stats[threadIdx.x] = 0.f;
}

// ---------------------------------------------------------------------------
// Kernel 2: gathered GEMM (16xCOUT tiles via V_WMMA_F32_16X16X4_F32) +
// atomic scatter-add. One wave per tile; each wave pins one kernel-offset k
// so the 32x64 weight slab stays resident in registers across tiles.
// ---------------------------------------------------------------------------
__global__ void __launch_bounds__(256)
spconv_wmma_kernel(const float* __restrict__ feats,
                   const float* __restrict__ W,
                   const int*   __restrict__ gidx,
                   const int*   __restrict__ sidx,
                   float*       __restrict__ out,
                   int P, int n_out) {
  const int lane = threadIdx.x & 31;
  const int wavesPerBlock = blockDim.x >> 5;
  const int gwid   = blockIdx.x * wavesPerBlock + (threadIdx.x >> 5);
  const int nwaves = gridDim.x * wavesPerBlock;

  const int k        = gwid & (KVOL - 1);   // fixed kernel offset per wave
  const int waveInK  = gwid >> 2;
  const int wavesPerK = nwaves >> 2;
  const int tilesPerK = (P + 15) >> 4;

  const int half = lane >> 4;   // 0: K even pair, 1: K odd pair (A/B operand split)
  const int m    = lane & 15;   // row (A) / column-in-block (B, C/D)

  // Preload B operands for all 4 N-blocks x 8 K-steps:
  // lane(half,m) needs W[k][4*kk+2*half][nb*16+m] and W[k][4*kk+2*half+1][nb*16+m]
  f32x2 b[4][8];
  const float* Wk = W + k * CIN * COUT;
#pragma unroll
  for (int nb = 0; nb < 4; ++nb) {
#pragma unroll
    for (int kk = 0; kk < 8; ++kk) {
      const int kcol = 4 * kk + 2 * half;
      const int col  = nb * 16 + m;
      f32x2 t;
      t.x = Wk[kcol * COUT + col];
      t.y = Wk[(kcol + 1) * COUT + col];
      b[nb][kk] = t;
    }
  }

  const int* gk = gidx + k * P;
  const int* sk = sidx + k * P;

  for (int tile = waveInK; tile < tilesPerK; tile += wavesPerK) {
    const int r0 = tile << 4;

    // Gather A tile: lane(half,m) loads row gidx[r0+m], float2 at column
    // 4*kk + 2*half  ->  exactly the 16x4-per-step WMMA A layout.
    const int ridx = r0 + m;
    const int row  = (ridx < P) ? gk[ridx] : 0;
    const float* arow = feats + (size_t)row * CIN;
    f32x2 a[8];
#pragma unroll
    for (int kk = 0; kk < 8; ++kk)
      a[kk] = *(const f32x2*)(arow + 4 * kk + 2 * half);

    f32x8 acc[4] = {{}, {}, {}, {}};

#pragma unroll
    for (int kk = 0; kk < 8; ++kk) {
#pragma unroll
      for (int nb = 0; nb < 4; ++nb) {
        acc[nb] = __builtin_amdgcn_wmma_f32_16x16x4_f32(
            /*neg_a=*/false, a[kk],
            /*neg_b=*/false, b[nb][kk],
            /*c_mod=*/(short)0, acc[nb],
            /*reuse_a=*/false, /*reuse_b=*/false);
      }
    }

    // Scatter: C/D VGPR j holds rows M=j (lanes 0-15) / M=j+8 (lanes 16-31),
    // column = nb*16 + m. Padding entries carry sidx == n_out -> skipped.
#pragma unroll
    for (int j = 0; j < 8; ++j) {
      const int midx = r0 + j + 8 * half;
      if (midx < P) {
        const int s = sk[midx];
        if (s < n_out) {
          float* orow = out + (size_t)s * COUT + m;
#pragma unroll
          for (int nb = 0; nb < 4; ++nb) {
            __hip_atomic_fetch_add(orow + nb * 16, acc[nb][j],
                                   __ATOMIC_RELAXED, __HIP_MEMORY_SCOPE_AGENT);
          }
        }
      }
    }
  }
}

// ---------------------------------------------------------------------------
// Kernel 3: per-channel sum / sum-of-squares. Grid stride is a multiple of
// 64, so each thread's channel (threadIdx.x % 64) is invariant.
// ---------------------------------------------------------------------------
__global__ void __launch_bounds__(256)
reduce_stats_kernel(const float* __restrict__ out, int n_out,
                    float* __restrict__ stats) {
  __shared__ float ssum[256];
  __shared__ float ssq[256];
  const long long total  = (long long)n_out * COUT;
  const long long stride = (long long)gridDim.x * blockDim.x;
  float s = 0.f, q = 0.f;
  for (long long i = (long long)blockIdx.x * blockDim.x + threadIdx.x;
       i < total; i += stride) {
    const float v = out[i];
    s += v;
    q += v * v;
  }
  ssum[threadIdx.x] = s;
  ssq[threadIdx.x]  = q;
  __syncthreads();
  if (threadIdx.x < COUT) {
    float ts = ssum[threadIdx.x] + ssum[threadIdx.x + 64] +
               ssum[threadIdx.x + 128] + ssum[threadIdx.x + 192];
    float tq = ssq[threadIdx.x] + ssq[threadIdx.x + 64] +
               ssq[threadIdx.x + 128] + ssq[threadIdx.x + 192];
    __hip_atomic_fetch_add(&stats[threadIdx.x], ts,
                           __ATOMIC_RELAXED, __HIP_MEMORY_SCOPE_AGENT);
    __hip_atomic_fetch_add(&stats[COUT + threadIdx.x], tq,
                           __ATOMIC_RELAXED, __HIP_MEMORY_SCOPE_AGENT);
  }
}

// ---------------------------------------------------------------------------
// Kernel 4: fold mean/var/gamma/beta into per-channel scale & shift.
// ---------------------------------------------------------------------------
__global__ void bn_params_kernel(const float* __restrict__ stats,
                                 const float* __restrict__ gamma,
                                 const float* __restrict__ beta,
                                 int n_out, float* __restrict__ params) {
  const int c = threadIdx.x;
  if (c < COUT) {
    const float inv_n = 1.f / (float)n_out;
    const float mean  = stats[c] * inv_n;
    const float var   = stats[COUT + c] * inv_n - mean * mean;
    const float sc    = gamma[c] * rsqrtf(var + BN_EPS);
    params[c]        = sc;
    params[COUT + c] = beta[c] - mean * sc;
  }
}

// ---------------------------------------------------------------------------
// Kernel 5: in-place normalize + ReLU, float4-vectorized.
// ---------------------------------------------------------------------------
__global__ void __launch_bounds__(256)
bn_apply_kernel(float* __restrict__ out, long long n4,
                const float* __restrict__ params) {
  const long long stride = (long long)gridDim.x * blockDim.x;
  for (long long i = (long long)blockIdx.x * blockDim.x + threadIdx.x;
       i < n4; i += stride) {
    f32x4 v = ((f32x4*)out)[i];
    const int cbase = (int)((i * 4) & (COUT - 1));
#pragma unroll
    for (int t = 0; t < 4; ++t) {
      const float r = v[t] * params[cbase + t] + params[COUT + cbase + t];
      v[t] = fmaxf(r, 0.f);
    }
    ((f32x4*)out)[i] = v;
  }
}

// ---------------------------------------------------------------------------
extern "C" void kernel_launch(void* const* d_in, const int* in_sizes, int n_in,
                              void* d_out, int out_size, void* d_ws,
                              size_t ws_size, hipStream_t stream) {
  const float* feats = (const float*)d_in[0];
  const float* W     = (const float*)d_in[1];
  const float* gamma = (const float*)d_in[2];
  const float* beta  = (const float*)d_in[3];
  const int*   gidx  = (const int*)d_in[4];
  const int*   sidx  = (const int*)d_in[5];
  // d_in[6] is n_out as a device scalar; derive on host instead (graph-safe):
  const int n_out = out_size / COUT;
  const int P     = in_sizes[4] / KVOL;

  float* out    = (float*)d_out;
  float* stats  = (float*)d_ws;      // [0..63] sum, [64..127] sumsq
  float* params = stats + 2 * COUT;  // [0..63] scale, [64..127] shift

  zero_kernel<<<2048, 256, 0, stream>>>(out, out_size, stats);
  spconv_wmma_kernel<<<1024, 256, 0, stream>>>(feats, W, gidx, sidx, out,
                                               P, n_out);
  reduce_stats_kernel<<<2048, 256, 0, stream>>>(out, n_out, stats);
  bn_params_kernel<<<1, 64, 0, stream>>>(stats, gamma, beta, n_out, params);
  bn_apply_kernel<<<2048, 256, 0, stream>>>(out, (long long)out_size / 4,
                                            params);
}